// DCRNNModel_nextTimePred_60696477827521
// MI455X (gfx1250) — compile-verified
//
#include <hip/hip_runtime.h>
#include <hip/hip_bf16.h>
#include <math.h>

// ---------------- problem constants ----------------
constexpr int cB   = 32;
constexpr int cTIN = 12;
constexpr int cTOUT= 12;
constexpr int cN   = 200;
constexpr int cDIN = 2;
constexpr int cDOUT= 1;
constexpr int cH   = 128;
constexpr int cDH  = 64;     // H/2
constexpr int cTOPK= 20;
constexpr float cInvSqrtH = 0.088388347648318447f; // 1/sqrt(128)

// padded activation geometry (diffusion/proj operands)
constexpr int cROWP = 208;   // 13 * 16 row tiles
constexpr int cLDX  = 256;   // feature stride (max Dc = 256)

typedef __attribute__((ext_vector_type(2))) float v2f;
typedef __attribute__((ext_vector_type(8))) float v8f;

// ---------------- workspace layout (floats) ----------------
constexpr size_t PADBUF    = (size_t)cB * cROWP * cLDX;          // 1,703,936
constexpr size_t OFF_OUTS  = 0;                                  // (T,N,B,H)
constexpr size_t OFF_POOL  = OFF_OUTS + (size_t)cTIN*cN*cB*cH;
constexpr size_t OFF_KEYM  = OFF_POOL + (size_t)cB*cN*cH;
constexpr size_t OFF_QM    = OFF_KEYM + (size_t)cB*cN*cDH;
constexpr size_t OFF_ADJ   = OFF_QM   + (size_t)cB*cN*cDH;
constexpr size_t OFF_DR    = OFF_ADJ  + (size_t)cB*cN*cN;
constexpr size_t OFF_DC    = OFF_DR   + (size_t)cB*cN;
constexpr size_t OFF_S1    = OFF_DC   + (size_t)cB*cN;
constexpr size_t OFF_S2    = OFF_S1   + (size_t)cB*cN*cN;
constexpr size_t OFF_HE0   = OFF_S2   + (size_t)cB*cN*cN;
constexpr size_t OFF_HE1   = OFF_HE0  + (size_t)cB*cN*cH;
constexpr size_t OFF_CAT   = OFF_HE1  + (size_t)cB*cN*cH;        // padded
constexpr size_t OFF_X1A   = OFF_CAT  + PADBUF;                  // padded
constexpr size_t OFF_X2A   = OFF_X1A  + PADBUF;
constexpr size_t OFF_X1B   = OFF_X2A  + PADBUF;
constexpr size_t OFF_X2B   = OFF_X1B  + PADBUF;
constexpr size_t OFF_RU    = OFF_X2B  + PADBUF;                  // compact (B*N, 2H)
constexpr size_t OFF_CBUF  = OFF_RU   + (size_t)cB*cN*2*cH;      // compact (B*N, H)
constexpr size_t OFF_XDEC  = OFF_CBUF + (size_t)cB*cN*cH;        // (B*N, 1)
constexpr size_t OFF_WHHT  = OFF_XDEC + (size_t)cB*cN;           // (H, 3H)

__device__ __forceinline__ float sigf(float x) { return 1.f / (1.f + expf(-x)); }

// ---------------- zero ----------------
__global__ void zero_kernel(float* p, size_t n) {
  size_t i = (size_t)blockIdx.x * blockDim.x + threadIdx.x;
  if (i < n) p[i] = 0.f;
}

// ---------------- Whh (3H,H) -> WhhT (H,3H) for coalesced GRU reads ----------------
__global__ void transpose_whh_kernel(const float* __restrict__ Whh, float* __restrict__ WhhT) {
  int idx = blockIdx.x * blockDim.x + threadIdx.x;
  if (idx >= 3 * cH * cH) return;
  int j = idx / cH, k = idx % cH;
  WhhT[(size_t)k * (3 * cH) + j] = Whh[idx];
}

// ---------------- GRU over T_IN, one block per (n,b) row ----------------
__global__ void gru_kernel(const float* __restrict__ enc,
                           const float* __restrict__ Wih, const float* __restrict__ WhhT,
                           const float* __restrict__ bih, const float* __restrict__ bhh,
                           float* __restrict__ outs) {
  int row = blockIdx.x;           // n*B + b
  int n = row / cB, b = row % cB;
  int j = threadIdx.x;            // 0..127 (hidden unit)
  __shared__ float h[cH];
  h[j] = 0.f;
  __syncthreads();
  for (int t = 0; t < cTIN; ++t) {
    const float x0 = enc[(((size_t)b*cTIN + t)*cN + n)*cDIN + 0];
    const float x1 = enc[(((size_t)b*cTIN + t)*cN + n)*cDIN + 1];
    float ir = Wih[(size_t)j*cDIN]*x0       + Wih[(size_t)j*cDIN+1]*x1       + bih[j];
    float iz = Wih[(size_t)(cH+j)*cDIN]*x0  + Wih[(size_t)(cH+j)*cDIN+1]*x1  + bih[cH+j];
    float ig = Wih[(size_t)(2*cH+j)*cDIN]*x0+ Wih[(size_t)(2*cH+j)*cDIN+1]*x1+ bih[2*cH+j];
    float hr = bhh[j], hz = bhh[cH+j], hg = bhh[2*cH+j];
    for (int k = 0; k < cH; ++k) {
      float hv = h[k];
      const float* wt = &WhhT[(size_t)k * (3 * cH)];
      hr = fmaf(wt[j],        hv, hr);
      hz = fmaf(wt[cH + j],   hv, hz);
      hg = fmaf(wt[2*cH + j], hv, hg);
    }
    float r  = sigf(ir + hr);
    float z  = sigf(iz + hz);
    float nn = tanhf(ig + r * hg);
    float hnew = (1.f - z) * nn + z * h[j];
    __syncthreads();
    h[j] = hnew;
    __syncthreads();
    outs[(((size_t)t*cN + n)*cB + b)*cH + j] = hnew;
  }
}

// ---------------- temporal attention pooling ----------------
__global__ void attn_pool_kernel(const float* __restrict__ outs,
                                 const float* __restrict__ attW, const float* __restrict__ attb,
                                 const float* __restrict__ wlen, float* __restrict__ pooled) {
  int row = blockIdx.x;           // n*B + b
  int n = row / cB, b = row % cB;
  int j = threadIdx.x;            // 0..127
  __shared__ float red[cDH];
  __shared__ float logits[cTIN];
  for (int t = 0; t < cTIN; ++t) {
    if (j < cDH) {
      const float* ov = &outs[(((size_t)t*cN + n)*cB + b)*cH];
      float acc = attb[j];
      const float* w = &attW[(size_t)j*cH];
      for (int k = 0; k < cH; ++k) acc = fmaf(w[k], ov[k], acc);
      acc = fmaxf(acc, 0.f);
      red[j] = acc * wlen[j];
    }
    __syncthreads();
    if (j == 0) {
      float s = 0.f;
      for (int d = 0; d < cDH; ++d) s += red[d];
      logits[t] = s;
    }
    __syncthreads();
  }
  float mx = logits[0];
  for (int t = 1; t < cTIN; ++t) mx = fmaxf(mx, logits[t]);
  float w[cTIN], den = 0.f;
  for (int t = 0; t < cTIN; ++t) { w[t] = expf(logits[t] - mx); den += w[t]; }
  float inv = 1.f / den, acc = 0.f;
  for (int t = 0; t < cTIN; ++t)
    acc = fmaf(w[t] * inv, outs[(((size_t)t*cN + n)*cB + b)*cH + j], acc);
  pooled[((size_t)b*cN + n)*cH + j] = acc;
}

// ---------------- key / query projections ----------------
__global__ void keyquery_kernel(const float* __restrict__ pooled,
                                const float* __restrict__ wkey, const float* __restrict__ wq,
                                float* __restrict__ keym, float* __restrict__ qm) {
  int row = blockIdx.x;           // n*B + b
  int n = row / cB, b = row % cB;
  int d = threadIdx.x;            // 0..63
  const float* p = &pooled[((size_t)b*cN + n)*cH];
  float ak = 0.f, aq = 0.f;
  for (int k = 0; k < cH; ++k) {
    float pv = p[k];
    ak = fmaf(pv, wkey[(size_t)k*cDH + d], ak);
    aq = fmaf(pv, wq  [(size_t)k*cDH + d], aq);
  }
  keym[((size_t)b*cN + n)*cDH + d] = ak;
  qm  [((size_t)b*cN + n)*cDH + d] = aq;
}

// ---------------- attention row: softmax, top-k threshold, adjacency, sums ----------------
__global__ void attn_adj_kernel(const float* __restrict__ keym, const float* __restrict__ qm,
                                float* __restrict__ adj, float* __restrict__ dr, float* __restrict__ dc) {
  int n = blockIdx.x, b = blockIdx.y;
  int m = threadIdx.x;            // 0..255, valid < 200
  __shared__ float sred[256];
  __shared__ float vals[256];
  __shared__ float spv[256];
  __shared__ int   spi[256];
  __shared__ float kthS;

  float v = -1e30f;
  if (m < cN) {
    const float* kp = &keym[((size_t)b*cN + n)*cDH];
    const float* qp = &qm  [((size_t)b*cN + m)*cDH];
    float acc = 0.f;
    for (int d = 0; d < cDH; ++d) acc = fmaf(kp[d], qp[d], acc);
    v = acc * cInvSqrtH;
  }
  sred[m] = v;
  __syncthreads();
  for (int s = 128; s > 0; s >>= 1) {
    if (m < s) sred[m] = fmaxf(sred[m], sred[m + s]);
    __syncthreads();
  }
  float mx = sred[0];
  __syncthreads();
  float e = (m < cN) ? expf(v - mx) : 0.f;
  sred[m] = e;
  __syncthreads();
  for (int s = 128; s > 0; s >>= 1) {
    if (m < s) sred[m] += sred[m + s];
    __syncthreads();
  }
  float a = e / sred[0];          // attn value (0 for m>=N)
  __syncthreads();

  // top-k threshold (20th largest) via iterative argmax
  vals[m] = (m < cN) ? a : -1e30f;
  __syncthreads();
  for (int it = 0; it < cTOPK; ++it) {
    spv[m] = vals[m]; spi[m] = m;
    __syncthreads();
    for (int s = 128; s > 0; s >>= 1) {
      if (m < s && spv[m + s] > spv[m]) { spv[m] = spv[m + s]; spi[m] = spi[m + s]; }
      __syncthreads();
    }
    if (m == 0) {
      if (it == cTOPK - 1) kthS = spv[0];
      vals[spi[0]] = -1e30f;
    }
    __syncthreads();
  }
  float kth = kthS;

  float adjv = 0.f;
  if (m < cN) {
    adjv = (a >= kth ? a : 0.f) + (m == n ? 1.f : 0.f);
    adj[(((size_t)b*cN + n))*cN + m] = adjv;
  }
  sred[m] = adjv;
  __syncthreads();
  for (int s = 128; s > 0; s >>= 1) {
    if (m < s) sred[m] += sred[m + s];
    __syncthreads();
  }
  if (m == 0) dr[(size_t)b*cN + n] = sred[0];
  if (m < cN) atomicAdd(&dc[(size_t)b*cN + m], adjv);
}

// ---------------- build supports s1 = rw(adj)^T, s2 = rw(adj^T)^T ----------------
__global__ void supports_kernel(const float* __restrict__ adj,
                                const float* __restrict__ dr, const float* __restrict__ dc,
                                float* __restrict__ s1, float* __restrict__ s2) {
  size_t idx = (size_t)blockIdx.x * blockDim.x + threadIdx.x;
  size_t total = (size_t)cB * cN * cN;
  if (idx >= total) return;
  int b = (int)(idx / ((size_t)cN * cN));
  int rem = (int)(idx % ((size_t)cN * cN));
  int i = rem / cN, j = rem % cN;
  float drj = dr[(size_t)b*cN + j];
  float dcj = dc[(size_t)b*cN + j];
  float ir = (drj == 0.f) ? 0.f : 1.f / drj;
  float ic = (dcj == 0.f) ? 0.f : 1.f / dcj;
  s1[idx] = adj[(((size_t)b*cN + j))*cN + i] * ir;   // rw(adj)^T
  s2[idx] = adj[(((size_t)b*cN + i))*cN + j] * ic;   // rw(adj^T)^T
}

// ---------------- WMMA f32 16x16x4 diffusion: OUT = alpha*(S@X) + beta*X0 ----------------
// X/X0/OUT are padded (cROWP rows, cLDX stride); no guards anywhere (uniform kernel).
__global__ __launch_bounds__(128)
void diffusion_kernel(const float* __restrict__ S, const float* __restrict__ X,
                      const float* __restrict__ X0, float* __restrict__ OUT,
                      float alpha, float beta) {
  __shared__ float sA[16 * cN];                 // 16 rows of S, 12.8 KB
  const int rt = blockIdx.x, cg = blockIdx.y, b = blockIdx.z;
  const int tid = threadIdx.x, lane = tid & 31, wid = tid >> 5;
  const int r0 = rt * 16;

  // async DMA stage of 16 S rows into LDS (b128 chunks); zero-fill row padding
  for (int c = tid; c < 16 * (cN / 4); c += 128) {    // 800 16B chunks
    int rr = r0 + c / (cN / 4);
    int cc = (c % (cN / 4)) * 4;
    int li = (c / (cN / 4)) * cN + cc;
    if (rr < cN) {
      const float* gp = &S[(((size_t)b * cN + rr)) * cN + cc];
      unsigned lo = (unsigned)(size_t)(&sA[li]);
      asm volatile("global_load_async_to_lds_b128 %0, %1, off"
                   :: "v"(lo), "v"(gp) : "memory");
    } else {
      sA[li] = 0.f; sA[li + 1] = 0.f; sA[li + 2] = 0.f; sA[li + 3] = 0.f;
    }
  }
  asm volatile("s_wait_asynccnt 0x0" ::: "memory");
  __syncthreads();

  const int c0   = cg * 64 + wid * 16;
  const int am   = lane & 15;
  const int koff = 2 * (lane >> 4);
  const int ncol = c0 + (lane & 15);
  const float* Xb = X + (size_t)b * cROWP * cLDX + ncol;
  v8f acc = {};
#pragma unroll 4
  for (int k0 = 0; k0 < cN; k0 += 4) {
    int kk = k0 + koff;
    v2f a, bb;
    a.x = sA[am * cN + kk];
    a.y = sA[am * cN + kk + 1];
    bb.x = Xb[(size_t)kk * cLDX];
    bb.y = Xb[(size_t)(kk + 1) * cLDX];
    acc = __builtin_amdgcn_wmma_f32_16x16x4_f32(false, a, false, bb, (short)0, acc, false, false);
  }
  const int hb = 8 * (lane >> 4);
  float* Ob = OUT + (size_t)b * cROWP * cLDX + ncol;
  const float* X0b = X0 ? (X0 + (size_t)b * cROWP * cLDX + ncol) : nullptr;
#pragma unroll
  for (int v = 0; v < 8; ++v) {
    int row = r0 + v + hb;                      // <= 207, inside padded buffer
    float val = alpha * acc[v];
    if (beta != 0.f) val = fmaf(beta, X0b[(size_t)row * cLDX], val);
    Ob[(size_t)row * cLDX] = val;
  }
}

// ---------------- WMMA f32 16x16x4 gconv projection: OUT = act(XM @ W + b) ----------------
// XM gathered on the fly from 5 padded buffers: feature f = d*5 + m; OUT compact (B,N,Dout).
__global__ __launch_bounds__(128)
void gconv_proj_kernel(const float* __restrict__ x0, const float* __restrict__ x1a,
                       const float* __restrict__ x2a, const float* __restrict__ x1b,
                       const float* __restrict__ x2b, int D,
                       const float* __restrict__ W, const float* __restrict__ bias,
                       int Dout, float* __restrict__ OUT, int act) {
  const int rt = blockIdx.x, cg = blockIdx.y, b = blockIdx.z;
  const int lane = threadIdx.x & 31, wid = threadIdx.x >> 5;
  const int r0 = rt * 16, c0 = cg * 64 + wid * 16;
  const float* xs[5] = {x0, x1a, x2a, x1b, x2b};
  const int KD = 5 * D;
  const int mrow = r0 + (lane & 15);            // <= 207, inside padded buffers
  const int koff = 2 * (lane >> 4);
  const int ncol = c0 + (lane & 15);            // < Dout (Dout multiple of 64)
  const size_t abase = ((size_t)b * cROWP + mrow) * cLDX;
  v8f acc = {};
  const int KDm = KD & ~3;                      // unguarded main K
#pragma unroll 2
  for (int k0 = 0; k0 < KDm; k0 += 4) {
    if ((k0 & 127) == 0 && k0 + 128 < KD)
      __builtin_prefetch(&W[(size_t)(k0 + 128) * Dout + ncol], 0, 1);  // global_prefetch_b8
    int f0 = k0 + koff, f1 = f0 + 1;
    v2f a, bb;
    a.x = xs[f0 % 5][abase + (size_t)(f0 / 5)];
    a.y = xs[f1 % 5][abase + (size_t)(f1 / 5)];
    bb.x = W[(size_t)f0 * Dout + ncol];
    bb.y = W[(size_t)f1 * Dout + ncol];
    acc = __builtin_amdgcn_wmma_f32_16x16x4_f32(false, a, false, bb, (short)0, acc, false, false);
  }
  if (KDm < KD) {                               // guarded tail (KD % 4 != 0)
    int f0 = KDm + koff, f1 = f0 + 1;
    v2f a, bb;
    a.x = (f0 < KD) ? xs[f0 % 5][abase + (size_t)(f0 / 5)] : 0.f;
    a.y = (f1 < KD) ? xs[f1 % 5][abase + (size_t)(f1 / 5)] : 0.f;
    bb.x = (f0 < KD) ? W[(size_t)f0 * Dout + ncol] : 0.f;
    bb.y = (f1 < KD) ? W[(size_t)f1 * Dout + ncol] : 0.f;
    acc = __builtin_amdgcn_wmma_f32_16x16x4_f32(false, a, false, bb, (short)0, acc, false, false);
  }
  const int hb = 8 * (lane >> 4);
#pragma unroll
  for (int v = 0; v < 8; ++v) {
    int row = r0 + v + hb;
    if (row < cN) {
      float val = acc[v] + bias[ncol];
      if (act == 1)      val = sigf(val);
      else if (act == 2) val = tanhf(val);
      OUT[(((size_t)b * cN + row)) * Dout + ncol] = val;
    }
  }
}

// ---------------- elementwise: concat(x, [r*]h) into padded layout ----------------
__global__ void concat_kernel(const float* __restrict__ x, int Dx,
                              const float* __restrict__ h, const float* __restrict__ ru,
                              float* __restrict__ out) {
  const int Dc = Dx + cH;
  size_t idx = (size_t)blockIdx.x * blockDim.x + threadIdx.x;
  if (idx >= (size_t)cB * cN * Dc) return;
  size_t bn = idx / Dc;
  int d = (int)(idx % Dc);
  int b = (int)(bn / cN), n = (int)(bn % cN);
  float v;
  if (d < Dx) v = x[bn * Dx + d];
  else {
    int j = d - Dx;
    float hv = h[bn * cH + j];
    if (ru) hv *= ru[bn * (2 * cH) + j];    // r gate (already sigmoid)
    v = hv;
  }
  out[((size_t)b * cROWP + n) * cLDX + d] = v;
}

// concat with encoder-input time slice: x index = ((b*T+t)*N+n)*2 + d
__global__ void concat_enc_kernel(const float* __restrict__ enc, int t,
                                  const float* __restrict__ h, const float* __restrict__ ru,
                                  float* __restrict__ out) {
  const int Dc = cDIN + cH;
  size_t idx = (size_t)blockIdx.x * blockDim.x + threadIdx.x;
  if (idx >= (size_t)cB * cN * Dc) return;
  size_t bn = idx / Dc;
  int d = (int)(idx % Dc);
  int b = (int)(bn / cN), n = (int)(bn % cN);
  float v;
  if (d < cDIN) v = enc[(((size_t)b*cTIN + t)*cN + n)*cDIN + d];
  else {
    int j = d - cDIN;
    float hv = h[bn * cH + j];
    if (ru) hv *= ru[bn * (2 * cH) + j];
    v = hv;
  }
  out[((size_t)b * cROWP + n) * cLDX + d] = v;
}

// ---------------- h = u*h + (1-u)*c ----------------
__global__ void update_kernel(float* __restrict__ h, const float* __restrict__ ru,
                              const float* __restrict__ c) {
  size_t idx = (size_t)blockIdx.x * blockDim.x + threadIdx.x;
  if (idx >= (size_t)cB * cN * cH) return;
  size_t bn = idx / cH;
  int j = (int)(idx % cH);
  float u = ru[bn * (2 * cH) + cH + j];
  h[idx] = u * h[idx] + (1.f - u) * c[idx];
}

// ---------------- decoder output projection ----------------
__global__ void projout_kernel(const float* __restrict__ h1, const float* __restrict__ pW,
                               const float* __restrict__ pb, float* __restrict__ dout,
                               float* __restrict__ xdec, int t) {
  size_t idx = (size_t)blockIdx.x * blockDim.x + threadIdx.x;
  if (idx >= (size_t)cB * cN) return;
  int b = (int)(idx / cN), n = (int)(idx % cN);
  const float* hp = &h1[idx * cH];
  float acc = pb[0];
  for (int k = 0; k < cH; ++k) acc = fmaf(hp[k], pW[k], acc);
  dout[(((size_t)t*cB + b))*cN + n] = acc;   // outputs (T_OUT, B, N*D_OUT)
  xdec[idx] = acc;                            // feedback
}

// ---------------- host launcher ----------------
extern "C" void kernel_launch(void* const* d_in, const int* in_sizes, int n_in,
                              void* d_out, int out_size, void* d_ws, size_t ws_size,
                              hipStream_t stream) {
  (void)in_sizes; (void)n_in; (void)out_size; (void)ws_size;
  const float* enc      = (const float*)d_in[0];
  const float* gru_Wih  = (const float*)d_in[2];
  const float* gru_Whh  = (const float*)d_in[3];
  const float* gru_bih  = (const float*)d_in[4];
  const float* gru_bhh  = (const float*)d_in[5];
  const float* att_W    = (const float*)d_in[6];
  const float* att_b    = (const float*)d_in[7];
  const float* w_len_C  = (const float*)d_in[8];
  const float* w_key    = (const float*)d_in[9];
  const float* w_query  = (const float*)d_in[10];
  const float* enc_Wg0  = (const float*)d_in[11];
  const float* enc_bg0  = (const float*)d_in[12];
  const float* enc_Wc0  = (const float*)d_in[13];
  const float* enc_bc0  = (const float*)d_in[14];
  const float* enc_Wg1  = (const float*)d_in[15];
  const float* enc_bg1  = (const float*)d_in[16];
  const float* enc_Wc1  = (const float*)d_in[17];
  const float* enc_bc1  = (const float*)d_in[18];
  const float* dec_Wg0  = (const float*)d_in[19];
  const float* dec_bg0  = (const float*)d_in[20];
  const float* dec_Wc0  = (const float*)d_in[21];
  const float* dec_bc0  = (const float*)d_in[22];
  const float* dec_Wg1  = (const float*)d_in[23];
  const float* dec_bg1  = (const float*)d_in[24];
  const float* dec_Wc1  = (const float*)d_in[25];
  const float* dec_bc1  = (const float*)d_in[26];
  const float* proj_W   = (const float*)d_in[27];
  const float* proj_b   = (const float*)d_in[28];

  float* ws    = (float*)d_ws;
  float* OUTS  = ws + OFF_OUTS;
  float* POOL  = ws + OFF_POOL;
  float* KEYM  = ws + OFF_KEYM;
  float* QM    = ws + OFF_QM;
  float* ADJ   = ws + OFF_ADJ;
  float* DR    = ws + OFF_DR;
  float* DC    = ws + OFF_DC;
  float* S1    = ws + OFF_S1;
  float* S2    = ws + OFF_S2;
  float* HE0   = ws + OFF_HE0;
  float* HE1   = ws + OFF_HE1;
  float* CAT   = ws + OFF_CAT;
  float* X1A   = ws + OFF_X1A;
  float* X2A   = ws + OFF_X2A;
  float* X1B   = ws + OFF_X1B;
  float* X2B   = ws + OFF_X2B;
  float* RU    = ws + OFF_RU;
  float* CBUF  = ws + OFF_CBUF;
  float* XDEC  = ws + OFF_XDEC;
  float* WHHT  = ws + OFF_WHHT;

  auto zero = [&](float* p, size_t n) {
    zero_kernel<<<(unsigned)((n + 255) / 256), 256, 0, stream>>>(p, n);
  };

  // ---- supports ----
  transpose_whh_kernel<<<(3 * cH * cH + 255) / 256, 256, 0, stream>>>(gru_Whh, WHHT);
  gru_kernel<<<cN * cB, cH, 0, stream>>>(enc, gru_Wih, WHHT, gru_bih, gru_bhh, OUTS);
  attn_pool_kernel<<<cN * cB, cH, 0, stream>>>(OUTS, att_W, att_b, w_len_C, POOL);
  keyquery_kernel<<<cN * cB, cDH, 0, stream>>>(POOL, w_key, w_query, KEYM, QM);
  zero(DC, (size_t)cB * cN);
  attn_adj_kernel<<<dim3(cN, cB), 256, 0, stream>>>(KEYM, QM, ADJ, DR, DC);
  supports_kernel<<<(unsigned)(((size_t)cB*cN*cN + 255) / 256), 256, 0, stream>>>(ADJ, DR, DC, S1, S2);

  // ---- DCGRU cell ----
  auto run_cell = [&](const float* xin, int Dx, bool enc_slice, int t, float* h,
                      const float* Wg, const float* bg, const float* Wc, const float* bc) {
    const int Dc = Dx + cH;
    const size_t ncat = (size_t)cB * cN * Dc;
    const dim3 dgrid(13, (unsigned)((Dc + 63) / 64), cB);
    // gate gconv
    if (enc_slice)
      concat_enc_kernel<<<(unsigned)((ncat + 255) / 256), 256, 0, stream>>>(xin, t, h, nullptr, CAT);
    else
      concat_kernel<<<(unsigned)((ncat + 255) / 256), 256, 0, stream>>>(xin, Dx, h, nullptr, CAT);
    diffusion_kernel<<<dgrid, 128, 0, stream>>>(S1, CAT, nullptr, X1A, 1.f, 0.f);
    diffusion_kernel<<<dgrid, 128, 0, stream>>>(S1, X1A, CAT,     X2A, 2.f, -1.f);
    diffusion_kernel<<<dgrid, 128, 0, stream>>>(S2, CAT, nullptr, X1B, 1.f, 0.f);
    diffusion_kernel<<<dgrid, 128, 0, stream>>>(S2, X1B, CAT,     X2B, 2.f, -1.f);
    gconv_proj_kernel<<<dim3(13, (2 * cH + 63) / 64, cB), 128, 0, stream>>>(
        CAT, X1A, X2A, X1B, X2B, Dc, Wg, bg, 2 * cH, RU, /*sigmoid*/1);
    // candidate gconv with r*h
    if (enc_slice)
      concat_enc_kernel<<<(unsigned)((ncat + 255) / 256), 256, 0, stream>>>(xin, t, h, RU, CAT);
    else
      concat_kernel<<<(unsigned)((ncat + 255) / 256), 256, 0, stream>>>(xin, Dx, h, RU, CAT);
    diffusion_kernel<<<dgrid, 128, 0, stream>>>(S1, CAT, nullptr, X1A, 1.f, 0.f);
    diffusion_kernel<<<dgrid, 128, 0, stream>>>(S1, X1A, CAT,     X2A, 2.f, -1.f);
    diffusion_kernel<<<dgrid, 128, 0, stream>>>(S2, CAT, nullptr, X1B, 1.f, 0.f);
    diffusion_kernel<<<dgrid, 128, 0, stream>>>(S2, X1B, CAT,     X2B, 2.f, -1.f);
    gconv_proj_kernel<<<dim3(13, (cH + 63) / 64, cB), 128, 0, stream>>>(
        CAT, X1A, X2A, X1B, X2B, Dc, Wc, bc, cH, CBUF, /*tanh*/2);
    update_kernel<<<(unsigned)(((size_t)cB*cN*cH + 255) / 256), 256, 0, stream>>>(h, RU, CBUF);
  };

  // ---- encoder ----
  zero(HE0, (size_t)cB * cN * cH);
  zero(HE1, (size_t)cB * cN * cH);
  for (int t = 0; t < cTIN; ++t) {
    run_cell(enc, cDIN, true, t, HE0, enc_Wg0, enc_bg0, enc_Wc0, enc_bc0);
    run_cell(HE0, cH, false, 0, HE1, enc_Wg1, enc_bg1, enc_Wc1, enc_bc1);
  }

  // ---- decoder (initial hiddens = final encoder hiddens, reused in place) ----
  zero(XDEC, (size_t)cB * cN * cDOUT);
  for (int t = 0; t < cTOUT; ++t) {
    run_cell(XDEC, cDOUT, false, 0, HE0, dec_Wg0, dec_bg0, dec_Wc0, dec_bc0);
    run_cell(HE0, cH, false, 0, HE1, dec_Wg1, dec_bg1, dec_Wc1, dec_bc1);
    projout_kernel<<<(unsigned)(((size_t)cB*cN + 255) / 256), 256, 0, stream>>>(
        HE1, proj_W, proj_b, (float*)d_out, XDEC, t);
  }
}